// HMM_4956392260151
// MI455X (gfx1250) — compile-verified
//
#include <hip/hip_runtime.h>
#include <hip/hip_bf16.h>
#include <math.h>

// HMM forward-backward posterior, B=256, T=1024, S=64.
//
// Key identity: emissions are scalar-per-(b,t) (log_emission_probs is (S,)
// indexed by obs), so they shift all states uniformly and cancel in the final
// log_softmax over states. Output = log_softmax_s(f[t,s] + g[t,s]) where
// f[t] = pi (x) A^t and g[t] = A^(T-1-t) (x) 0 in the logsumexp semiring —
// independent of batch/observations. So:
//   kernel 1: doubling scan (depth 10) computing f, g with WMMA f32 16x16x4
//   kernel 2: bandwidth-bound broadcast write of 64 MiB output.

typedef __attribute__((ext_vector_type(2))) float v2f;
typedef __attribute__((ext_vector_type(8))) float v8f;

#define HMM_B 256
#define HMM_T 1024
#define HMM_S 64

__global__ __launch_bounds__(256) void hmm_scan_kernel(
    const float* __restrict__ Alog,   // (64,64) log transition, row = from
    const float* __restrict__ pi,     // (64,)  log initial
    float* __restrict__ Fg,           // ws: (T,64)  f[t]
    float* __restrict__ Hg,           // ws: (T,64)  H[u] = g[T-1-u]
    float* __restrict__ LG)           // ws: (T,64)  log_softmax(f+g)
{
  __shared__ float Mlog[64][64];      // current power M_k (log domain)
  __shared__ float EmF[64][64];       // EmF[k][j] = exp(Mlog[k][j]-mu[j])
  __shared__ float EmB[64][64];       // EmB[j][i] = exp(Mlog[i][j]-nu[i])
  __shared__ float Pf[16][64];        // staged exp rows (forward)
  __shared__ float Ph[16][64];        // staged exp rows (backward)
  __shared__ float mu[64], nu[64], mF[16], mH[16];

  const int tid  = threadIdx.x;
  const int w    = tid >> 5;          // wave id (8 waves)
  const int lane = tid & 31;
  const int half = lane >> 4;
  const int rr   = lane & 15;

  // init: M_0 = A; f[0] = pi; g[T-1] = 0
  for (int e = tid; e < 4096; e += 256) Mlog[e >> 6][e & 63] = Alog[e];
  if (tid < 64) { Fg[tid] = pi[tid]; Hg[tid] = 0.0f; }
  __syncthreads();

  for (int k = 0; k < 10; ++k) {
    const int R = 1 << k;             // rows available; produces rows [R, 2R)

    // ---- column/row scales and exp'd matrices for M_k ----
    if (tid < 128) {
      const int i = tid & 63;
      float mx = -INFINITY;
      if (tid < 64) {                 // nu_i = row max
        for (int j = 0; j < 64; ++j) mx = fmaxf(mx, Mlog[i][j]);
        nu[i] = mx;
      } else {                        // mu_j = col max
        for (int kk = 0; kk < 64; ++kk) mx = fmaxf(mx, Mlog[kk][i]);
        mu[i] = mx;
      }
    }
    __syncthreads();
    for (int e = tid; e < 4096; e += 256) {
      const int i = e >> 6, j = e & 63;
      EmF[i][j] = expf(Mlog[i][j] - mu[j]);
      EmB[j][i] = expf(Mlog[i][j] - nu[i]);
    }
    __syncthreads();

    // ---- rows: f[R+r] = f[r] (x) M_k ; h[R+r] = h[r] (x) M_k^T ----
    for (int r0 = 0; r0 < R; r0 += 16) {
      const int valid = (R - r0 < 16) ? (R - r0) : 16;

      // phase A: stage p = exp(row - rowmax) into LDS (wave 0 only)
      if (tid < 32) {
        const int  rl  = tid & 15;
        const bool bwd = tid >= 16;
        const int  gr  = r0 + rl;
        float (*P)[64] = bwd ? Ph : Pf;
        float* mrow    = bwd ? mH : mF;
        if (gr < R) {
          const float* src = (bwd ? Hg : Fg) + (size_t)gr * 64;
          float mx = -INFINITY;
          for (int i = 0; i < 64; ++i) mx = fmaxf(mx, src[i]);
          for (int i = 0; i < 64; ++i) P[rl][i] = expf(src[i] - mx);
          mrow[rl] = mx;
        } else {
          for (int i = 0; i < 64; ++i) P[rl][i] = 0.0f;
          mrow[rl] = 0.0f;
        }
      }
      __syncthreads();

      // phase B: one 16x16 output tile per wave: (dir, col-tile)
      {
        const int dir = w >> 2;       // 0 = fwd, 1 = bwd
        const int nt  = w & 3;        // column tile of 16
        float (*P)[64] = dir ? Ph : Pf;
        float (*E)[64] = dir ? EmB : EmF;
        v8f acc = {0.f,0.f,0.f,0.f,0.f,0.f,0.f,0.f};
        const int c = nt * 16 + rr;
        #pragma unroll
        for (int kk = 0; kk < 16; ++kk) {
          const int k0 = kk * 4 + half * 2;   // per-ISA A/B K striping
          v2f a, b;
          a.x = P[rr][k0];  a.y = P[rr][k0 + 1];
          b.x = E[k0][c];   b.y = E[k0 + 1][c];
          acc = __builtin_amdgcn_wmma_f32_16x16x4_f32(
              false, a, false, b, (short)0, acc, false, false);
        }
        float*       dst  = dir ? Hg : Fg;
        const float* mrow = dir ? mH : mF;
        const float* scal = dir ? nu : mu;
        #pragma unroll
        for (int v = 0; v < 8; ++v) {
          const int rl = v + 8 * half;        // C/D row = VGPR + 8*lane-half
          if (rl < valid)
            dst[(size_t)(R + r0 + rl) * 64 + c] =
                mrow[rl] + scal[c] + logf(acc[v]);
        }
      }
      __syncthreads();
    }

    // ---- squaring: M_{k+1}[i][j] = nu_i + mu_j + log(EmB^T @ EmF) ----
    if (k < 9) {
      for (int task = w; task < 16; task += 8) {   // wave-uniform loop
        const int ti = task >> 2, tj = task & 3;
        v8f acc = {0.f,0.f,0.f,0.f,0.f,0.f,0.f,0.f};
        const int c  = tj * 16 + rr;
        const int rA = ti * 16 + rr;
        #pragma unroll
        for (int kk = 0; kk < 16; ++kk) {
          const int k0 = kk * 4 + half * 2;
          v2f a, b;
          a.x = EmB[k0][rA]; a.y = EmB[k0 + 1][rA];
          b.x = EmF[k0][c];  b.y = EmF[k0 + 1][c];
          acc = __builtin_amdgcn_wmma_f32_16x16x4_f32(
              false, a, false, b, (short)0, acc, false, false);
        }
        #pragma unroll
        for (int v = 0; v < 8; ++v) {
          const int rl = ti * 16 + v + 8 * half;
          Mlog[rl][c] = nu[rl] + mu[c] + logf(acc[v]);
        }
      }
      __syncthreads();
    }
  }

  // ---- LG[t] = log_softmax_s(f[t] + g[t]),  g[t] = H[T-1-t] ----
  for (int t = tid; t < HMM_T; t += 256) {
    const float* Ft = Fg + (size_t)t * 64;
    const float* Ht = Hg + (size_t)(HMM_T - 1 - t) * 64;
    float mx = -INFINITY;
    for (int i = 0; i < 64; ++i) mx = fmaxf(mx, Ft[i] + Ht[i]);
    float s = 0.0f;
    for (int i = 0; i < 64; ++i) s += expf(Ft[i] + Ht[i] - mx);
    const float lse = mx + logf(s);
    for (int i = 0; i < 64; ++i)
      LG[(size_t)t * 64 + i] = Ft[i] + Ht[i] - lse;
  }
}

// Bandwidth-bound broadcast: out[b,t,s] = LG[t,s]. 64 MiB of float4 stores;
// LG (256 KB) stays L2-resident.
__global__ __launch_bounds__(256) void hmm_bcast_kernel(
    const float* __restrict__ LG, float* __restrict__ out)
{
  const size_t i = (size_t)blockIdx.x * 256 + threadIdx.x;   // float4 index
  const float4* L4 = (const float4*)LG;
  float4* O4 = (float4*)out;
  O4[i] = L4[i & ((size_t)HMM_T * HMM_S / 4 - 1)];
}

extern "C" void kernel_launch(void* const* d_in, const int* in_sizes, int n_in,
                              void* d_out, int out_size, void* d_ws, size_t ws_size,
                              hipStream_t stream) {
  (void)in_sizes; (void)n_in; (void)out_size; (void)ws_size;
  // d_in[0] = observations (unused: emission terms cancel in log_softmax)
  const float* Alog = (const float*)d_in[1];   // (S,S)
  const float* pi   = (const float*)d_in[2];   // (S,)
  // d_in[3] = log_emission_probs (unused: cancels)

  float* ws = (float*)d_ws;
  float* Fg = ws;                               // T*S floats
  float* Hg = ws + (size_t)HMM_T * HMM_S;       // T*S floats
  float* LG = ws + (size_t)2 * HMM_T * HMM_S;   // T*S floats

  hmm_scan_kernel<<<1, 256, 0, stream>>>(Alog, pi, Fg, Hg, LG);

  const int n4 = HMM_B * HMM_T * HMM_S / 4;     // float4 elements
  hmm_bcast_kernel<<<n4 / 256, 256, 0, stream>>>(LG, (float*)d_out);
}